// DGLSage_4733053960603
// MI455X (gfx1250) — compile-verified
//
#include <hip/hip_runtime.h>

// GraphSAGE (3-layer, max-pool aggregator) for MI455X / gfx1250.
// GEMMs via V_WMMA_F32_16X16X4_F32 (full fp32 precision — GEMMs are not the
// bottleneck; the edge scatter-max is HBM/L2-atomic bound: ~1.6GB gather +
// 410M atomics per layer, hN (102MB) resident in the 192MB L2).
// Scatter-max via order-preserving float->u32 encoding + global_atomic_max_u32
// (deterministic: max is order-independent, encoding is a monotonic bijection).

typedef __attribute__((ext_vector_type(2))) float v2f;
typedef __attribute__((ext_vector_type(8))) float v8f;

#define P_POOL 512
#define ENC_NEG_INF 0x007FFFFFu   // enc(-inf): ~0xFF800000

__device__ __forceinline__ unsigned enc_f32(float f) {
  int i = __float_as_int(f);
  return (i >= 0) ? ((unsigned)i | 0x80000000u) : ~((unsigned)i);
}

__device__ __forceinline__ float dec_f32_zfix(unsigned u) {
  // decode + DGL semantics: nodes with no in-edges (still -inf) become 0
  int i = (u & 0x80000000u) ? (int)(u & 0x7FFFFFFFu) : (int)(~u);
  float f = __int_as_float(i);
  return (u == ENC_NEG_INF) ? 0.0f : f;
}

// ---------------------------------------------------------------------------
// Fill hN with encoded -inf
// ---------------------------------------------------------------------------
__global__ void fill_enc_neginf(unsigned* __restrict__ p, size_t n) {
  size_t i = (size_t)blockIdx.x * blockDim.x + threadIdx.x;
  if (i < n) p[i] = ENC_NEG_INF;
}

// ---------------------------------------------------------------------------
// One block per edge: atomic max of h[src[e], :] into hN[dst[e], :]
// src/dst are uniform per block -> scalar loads; atomics resolve in L2.
// ---------------------------------------------------------------------------
__global__ __launch_bounds__(256) void sage_scatter_max(
    const float* __restrict__ h, const int* __restrict__ src,
    const int* __restrict__ dst, unsigned* __restrict__ hN, int E) {
  int e = blockIdx.x;
  if (e >= E) return;
  int s = src[e];
  int d = dst[e];
  const float* hs = h + (size_t)s * P_POOL;
  unsigned* hd = hN + (size_t)d * P_POOL;
  for (int j = threadIdx.x; j < P_POOL; j += blockDim.x) {
    atomicMax(&hd[j], enc_f32(hs[j]));
  }
}

// ---------------------------------------------------------------------------
// WMMA fp32 GEMM: out[nrows, ncols] = Aeff[nrows, K0+P2] @ W[K0+P2, ncols]
//   Aeff[:, 0:K0]     = A   (row stride K0)
//   Aeff[:, K0:K0+P2] = dec(hN)  (row stride P2; skipped when P2 == 0)
// One wave -> 32x64 output strip: 2 A-frags x 4 B-frags = 8 wmma / K-step.
// Two separate K-loops (x part, hN part) keep the hot loop branch-free; the
// W pointer is carried across both (strength-reduced, no per-iter mul).
// 8 waves per block -> 256x64 block tile. ncols must be a multiple of 64;
// K0 and P2 must be multiples of 4 (true here: 128/256 and 0/512).
// ---------------------------------------------------------------------------
__global__ __launch_bounds__(256) void sage_wmma_gemm(
    const float* __restrict__ A, int K0,
    const unsigned* __restrict__ hN, int P2,
    const float* __restrict__ W,
    float* __restrict__ out, int nrows, int ncols, int relu) {
  const int lane = threadIdx.x & 31;
  const int wave = threadIdx.x >> 5;
  const int row0 = blockIdx.x * 256 + wave * 32;
  const int col0 = blockIdx.y * 64;
  const int mrow = lane & 15;        // A row within 16-tile / out col within 16-tile
  const int kb   = (lane >> 4) << 1; // lanes 0-15 -> K {0,1}, lanes 16-31 -> K {2,3}

  // clamp rows for addressing; stores are guarded below
  int r0 = row0 + mrow;
  int r1 = r0 + 16;
  const int rA0 = (r0 < nrows) ? r0 : (nrows - 1);
  const int rA1 = (r1 < nrows) ? r1 : (nrows - 1);
  const float* Ar0 = A + (size_t)rA0 * K0 + kb;
  const float* Ar1 = A + (size_t)rA1 * K0 + kb;

  // B walker: rows {kb, kb+1} of W at col (col0 + t*16 + mrow); +4 rows / step
  const float* Wp = W + (size_t)kb * ncols + (col0 + mrow);
  const size_t wstep = (size_t)4 * ncols;

  v8f acc[2][4];
#pragma unroll
  for (int m = 0; m < 2; ++m)
#pragma unroll
    for (int t = 0; t < 4; ++t) acc[m][t] = (v8f){};

  // ---- part 1: dense x columns [0, K0) ----
#pragma unroll 2
  for (int k = 0; k < K0; k += 4) {
    v2f a0, a1;
    a0.x = Ar0[k]; a0.y = Ar0[k + 1];
    a1.x = Ar1[k]; a1.y = Ar1[k + 1];
    v2f b[4];
#pragma unroll
    for (int t = 0; t < 4; ++t) {
      b[t].x = Wp[t * 16];
      b[t].y = Wp[(size_t)ncols + t * 16];
    }
    Wp += wstep;
#pragma unroll
    for (int t = 0; t < 4; ++t) {
      acc[0][t] = __builtin_amdgcn_wmma_f32_16x16x4_f32(false, a0, false, b[t], (short)0, acc[0][t], false, false);
      acc[1][t] = __builtin_amdgcn_wmma_f32_16x16x4_f32(false, a1, false, b[t], (short)0, acc[1][t], false, false);
    }
  }

  // ---- part 2: aggregated-neighbor columns [K0, K0+P2), decoded from hN ----
  if (P2 > 0) {
    const unsigned* Hr0 = hN + (size_t)rA0 * P2 + kb;
    const unsigned* Hr1 = hN + (size_t)rA1 * P2 + kb;
#pragma unroll 2
    for (int k = 0; k < P2; k += 4) {
      v2f a0, a1;
      a0.x = dec_f32_zfix(Hr0[k]); a0.y = dec_f32_zfix(Hr0[k + 1]);
      a1.x = dec_f32_zfix(Hr1[k]); a1.y = dec_f32_zfix(Hr1[k + 1]);
      v2f b[4];
#pragma unroll
      for (int t = 0; t < 4; ++t) {
        b[t].x = Wp[t * 16];
        b[t].y = Wp[(size_t)ncols + t * 16];
      }
      Wp += wstep;
#pragma unroll
      for (int t = 0; t < 4; ++t) {
        acc[0][t] = __builtin_amdgcn_wmma_f32_16x16x4_f32(false, a0, false, b[t], (short)0, acc[0][t], false, false);
        acc[1][t] = __builtin_amdgcn_wmma_f32_16x16x4_f32(false, a1, false, b[t], (short)0, acc[1][t], false, false);
      }
    }
  }

  // D layout: VGPR vr -> M = vr + 8*(lane>=16), N = lane&15
  const int mhi = (lane >> 4) * 8;
#pragma unroll
  for (int m = 0; m < 2; ++m) {
#pragma unroll
    for (int t = 0; t < 4; ++t) {
#pragma unroll
      for (int vr = 0; vr < 8; ++vr) {
        int orow = row0 + m * 16 + mhi + vr;
        if (orow < nrows) {
          float v = acc[m][t][vr];
          if (relu) v = fmaxf(v, 0.0f);
          out[(size_t)orow * ncols + (col0 + t * 16 + mrow)] = v;
        }
      }
    }
  }
}

// ---------------------------------------------------------------------------
extern "C" void kernel_launch(void* const* d_in, const int* in_sizes, int n_in,
                              void* d_out, int out_size, void* d_ws, size_t ws_size,
                              hipStream_t stream) {
  (void)n_in; (void)out_size; (void)ws_size;

  const float* x     = (const float*)d_in[0];
  const int*   srcI  = (const int*)d_in[1];
  const int*   dstI  = (const int*)d_in[2];
  const float* aggW0 = (const float*)d_in[3];
  const float* aggW1 = (const float*)d_in[4];
  const float* aggW2 = (const float*)d_in[5];
  const float* linW0 = (const float*)d_in[6];
  const float* linW1 = (const float*)d_in[7];
  const float* linW2 = (const float*)d_in[8];
  float* out = (float*)d_out;

  const int N = 50000, D = 128, H = 256, P = 512, O = 64;
  const int E = in_sizes[1];
  const size_t NP = (size_t)N * P;
  const size_t NH = (size_t)N * H;

  // workspace: h[N,P] fp32 | hN[N,P] enc-u32 | x1[N,H] | x2[N,H]  (~307 MB)
  char* ws = (char*)d_ws;
  float*    h  = (float*)ws;
  unsigned* hN = (unsigned*)(ws + NP * sizeof(float));
  float*    x1 = (float*)(ws + 2 * NP * sizeof(float));
  float*    x2 = (float*)(ws + 2 * NP * sizeof(float) + NH * sizeof(float));

  dim3 blk(256);
  auto ggrid = [](int rows, int cols) {
    return dim3((unsigned)((rows + 255) / 256), (unsigned)(cols / 64));
  };
  dim3 fillg((unsigned)((NP + 255) / 256));
  dim3 edgeg((unsigned)E);

  // ---- layer 0: x[N,128] ----
  sage_wmma_gemm<<<ggrid(N, P), blk, 0, stream>>>(x, D, nullptr, 0, aggW0, h, N, P, 0);
  fill_enc_neginf<<<fillg, blk, 0, stream>>>(hN, NP);
  sage_scatter_max<<<edgeg, blk, 0, stream>>>(h, srcI, dstI, hN, E);
  sage_wmma_gemm<<<ggrid(N, H), blk, 0, stream>>>(x, D, hN, P, linW0, x1, N, H, 1);

  // ---- layer 1: x1[N,256] ----
  sage_wmma_gemm<<<ggrid(N, P), blk, 0, stream>>>(x1, H, nullptr, 0, aggW1, h, N, P, 0);
  fill_enc_neginf<<<fillg, blk, 0, stream>>>(hN, NP);
  sage_scatter_max<<<edgeg, blk, 0, stream>>>(h, srcI, dstI, hN, E);
  sage_wmma_gemm<<<ggrid(N, H), blk, 0, stream>>>(x1, H, hN, P, linW1, x2, N, H, 1);

  // ---- layer 2 (output, no relu): x2[N,256] -> out[N,64] ----
  sage_wmma_gemm<<<ggrid(N, P), blk, 0, stream>>>(x2, H, nullptr, 0, aggW2, h, N, P, 0);
  fill_enc_neginf<<<fillg, blk, 0, stream>>>(hN, NP);
  sage_scatter_max<<<edgeg, blk, 0, stream>>>(h, srcI, dstI, hN, E);
  sage_wmma_gemm<<<ggrid(N, O), blk, 0, stream>>>(x2, H, hN, P, linW2, out, N, O, 0);
}